// GCNConv_23802708754517
// MI455X (gfx1250) — compile-verified
//
#include <hip/hip_runtime.h>

typedef __attribute__((ext_vector_type(2))) float v2f;
typedef __attribute__((ext_vector_type(8))) float v8f;

#define CDIM 128

// ---------------- init kernels ----------------
__global__ void zero_out_kernel(float* __restrict__ out, int n) {
    int i = blockIdx.x * blockDim.x + threadIdx.x;
    if (i < n) out[i] = 0.0f;
}

__global__ void init_deg_kernel(float* __restrict__ deg, int n) {
    int i = blockIdx.x * blockDim.x + threadIdx.x;
    if (i < n) deg[i] = 1.0f;   // self-loop contributes 1 to every node's degree
}

// ---------------- degree count on col ----------------
__global__ void degree_kernel(const long long* __restrict__ col,
                              float* __restrict__ deg, int E) {
    int e = blockIdx.x * blockDim.x + threadIdx.x;
    if (e < E) atomicAdd(&deg[(int)col[e]], 1.0f);
}

// ---------------- deg -> deg^-1/2 (in place) ----------------
__global__ void rsqrt_kernel(float* __restrict__ deg, int n) {
    int i = blockIdx.x * blockDim.x + threadIdx.x;
    if (i < n) deg[i] = rsqrtf(deg[i]);   // deg >= 1 always (self loops)
}

// ---------------- h = x @ W via fp32 WMMA ----------------
// One wave computes 16 rows x 128 cols of h.
// A (16x4 f32) per-lane layout: lanes 0-15 -> M=lane, K={0,1}; lanes 16-31 -> M=lane-16, K={2,3}
// B (4x16 f32) per-lane layout: lanes 0-15 -> N=lane, K={0,1}; lanes 16-31 -> N=lane-16, K={2,3}
// C/D (16x16 f32): VGPR v: lanes 0-15 -> M=v, lanes 16-31 -> M=v+8, N = lane&15
__global__ void __launch_bounds__(128)
gemm_wmma_kernel(const float* __restrict__ x, const float* __restrict__ W,
                 float* __restrict__ h, int N) {
    const int lane    = threadIdx.x & 31;
    const int wave    = threadIdx.x >> 5;
    const int rowBase = (blockIdx.x * 4 + wave) * 16;
    if (rowBase >= N) return;              // wave-uniform: EXEC stays all-1s for WMMA

    const int nl = lane & 15;              // column / row within 16-tile
    const int hi = lane >> 4;              // half-wave select
    const int kb = hi * 2;                 // K sub-offset {0,2}

    v8f acc[8];
#pragma unroll
    for (int n = 0; n < 8; ++n) acc[n] = (v8f)0.0f;

    const float* __restrict__ arow = x + (size_t)(rowBase + nl) * CDIM;

    for (int k = 0; k < CDIM; k += 4) {
        v2f a;
        a.x = arow[k + kb];
        a.y = arow[k + kb + 1];
        const float* __restrict__ w0 = W + (size_t)(k + kb) * CDIM + nl;
        const float* __restrict__ w1 = W + (size_t)(k + kb + 1) * CDIM + nl;
#pragma unroll
        for (int n = 0; n < 8; ++n) {
            v2f b;
            b.x = w0[n * 16];
            b.y = w1[n * 16];
            acc[n] = __builtin_amdgcn_wmma_f32_16x16x4_f32(
                false, a, false, b, (short)0, acc[n], false, false);
        }
    }

#pragma unroll
    for (int n = 0; n < 8; ++n) {
#pragma unroll
        for (int v = 0; v < 8; ++v) {
            h[(size_t)(rowBase + v + hi * 8) * CDIM + n * 16 + nl] = acc[n][v];
        }
    }
}

// ---------------- edge gather + scatter-add ----------------
// One wave per edge; each lane handles 4 channels (float4 gather, 4 f32 atomics).
__global__ void __launch_bounds__(256)
scatter_kernel(const long long* __restrict__ rowi, const long long* __restrict__ coli,
               const float* __restrict__ dis, const float* __restrict__ h,
               float* __restrict__ out, int E) {
    const int lane = threadIdx.x & 31;
    const int e    = (int)((blockIdx.x * 256u + threadIdx.x) >> 5);
    if (e >= E) return;

    const int r = (int)rowi[e];            // wave-uniform -> scalar loads
    const int c = (int)coli[e];
    const float nrm = dis[r] * dis[c];

    const float4 hv = *(const float4*)(h + (size_t)c * CDIM + lane * 4);
    float* o = out + (size_t)r * CDIM + lane * 4;
    atomicAdd(o + 0, nrm * hv.x);
    atomicAdd(o + 1, nrm * hv.y);
    atomicAdd(o + 2, nrm * hv.z);
    atomicAdd(o + 3, nrm * hv.w);
}

// ---------------- self-loop term (stream-ordered after scatter; no atomics) ----------------
__global__ void selfloop_kernel(const float* __restrict__ dis, const float* __restrict__ h,
                                float* __restrict__ out, int total) {
    int i = blockIdx.x * blockDim.x + threadIdx.x;
    if (i < total) {
        const float d = dis[i >> 7];       // node index = i / 128
        out[i] += d * d * h[i];
    }
}

extern "C" void kernel_launch(void* const* d_in, const int* in_sizes, int n_in,
                              void* d_out, int out_size, void* d_ws, size_t ws_size,
                              hipStream_t stream) {
    const float*      x  = (const float*)d_in[0];
    const float*      W  = (const float*)d_in[1];
    const long long*  ei = (const long long*)d_in[2];   // int64 [2, E]

    const int N = in_sizes[0] / CDIM;
    const int E = in_sizes[2] / 2;
    const long long* rowi = ei;          // ei[0, :]
    const long long* coli = ei + E;      // ei[1, :]

    float* out = (float*)d_out;
    float* deg = (float*)d_ws;                              // N floats (becomes dis)
    float* h   = deg + (((size_t)N + 255) & ~(size_t)255);  // N*128 floats

    const int total = N * CDIM;

    zero_out_kernel<<<(total + 255) / 256, 256, 0, stream>>>(out, total);
    init_deg_kernel<<<(N + 255) / 256, 256, 0, stream>>>(deg, N);
    degree_kernel<<<(E + 255) / 256, 256, 0, stream>>>(coli, deg, E);
    rsqrt_kernel<<<(N + 255) / 256, 256, 0, stream>>>(deg, N);

    const int rowTiles = (N + 15) / 16;             // waves needed
    gemm_wmma_kernel<<<(rowTiles + 3) / 4, 128, 0, stream>>>(x, W, h, N);

    scatter_kernel<<<(E + 7) / 8, 256, 0, stream>>>(rowi, coli, deg, h, out, E);
    selfloop_kernel<<<(total + 255) / 256, 256, 0, stream>>>(deg, h, out, total);
}